// InstanceRecognizerReconstructor_49469433315678
// MI455X (gfx1250) — compile-verified
//
#include <hip/hip_runtime.h>

// ---------------------------------------------------------------------------
// out[b,s,y,x] = (P'_s · sim_s[b] · Q'_s)[y,x]   with normalization folded
// into P',Q' as reciprocal coverage counts (separable: count = ny(y)*nx(x)).
// Implemented with V_WMMA_F32_16X16X4_F32 (wave32, 16x16 tiles, K padded to 16).
// ---------------------------------------------------------------------------

typedef __attribute__((ext_vector_type(2))) float v2f;
typedef __attribute__((ext_vector_type(8))) float v8f;

#define H 96
#define W 128
#define STRIDE 8

__global__ __launch_bounds__(256) void winscatter_wmma_kernel(
    const float* __restrict__ sim0, const float* __restrict__ sim1,
    const float* __restrict__ sim2, const float* __restrict__ sim3,
    float* __restrict__ out, int B)
{
    __shared__ float Ps[96 * 16];      // P'  (96 x 16), zero outside [0,yw)
    __shared__ float Qs[16 * 128];     // Q'  (16 x 128), zero outside [0,xw)
    __shared__ float Apad[16 * 16];    // sim tile zero-padded to 16x16
    __shared__ float Ustage[8][16 * 16]; // per-wave U tile (row-major), D->B relayout

    const int b  = blockIdx.x;
    const int si = blockIdx.y;                 // scale index 0..3
    const int scale = (si == 0) ? 16 : (si == 1) ? 24 : (si == 2) ? 32 : 48;
    const int yw = (H - scale) / STRIDE + 1;   // 11,10,9,7
    const int xw = (W - scale) / STRIDE + 1;   // 15,14,13,11
    const float* sim = (si == 0) ? sim0 : (si == 1) ? sim1 : (si == 2) ? sim2 : sim3;

    const int tid = threadIdx.x;

    // ---- Build P' (96x16): Ps[y*16+wy] = (wy in [lo,hi]) ? 1/ny : 0 ----
    for (int idx = tid; idx < 96 * 16; idx += 256) {
        int y = idx >> 4, wy = idx & 15;
        int lo = y - scale + STRIDE; lo = (lo > 0) ? (lo / STRIDE) : 0; // ceil((y-s+1)/8) clamped
        int hi = y / STRIDE; if (hi > yw - 1) hi = yw - 1;
        int ny = hi - lo + 1;
        Ps[idx] = (wy >= lo && wy <= hi) ? (1.0f / (float)ny) : 0.0f;
    }
    // ---- Build Q' (16x128): Qs[wx*128+x] = (wx in [lo,hi]) ? 1/nx : 0 ----
    for (int idx = tid; idx < 16 * 128; idx += 256) {
        int wx = idx >> 7, x = idx & 127;
        int lo = x - scale + STRIDE; lo = (lo > 0) ? (lo / STRIDE) : 0;
        int hi = x / STRIDE; if (hi > xw - 1) hi = xw - 1;
        int nx = hi - lo + 1;
        Qs[idx] = (wx >= lo && wx <= hi) ? (1.0f / (float)nx) : 0.0f;
    }
    // ---- Load sim tile, zero-padded to 16x16 ----
    {
        int idx = tid;                 // 256 threads, 256 elements
        int wy = idx >> 4, wx = idx & 15;
        float v = 0.0f;
        if (wy < yw && wx < xw)
            v = sim[(size_t)b * yw * xw + (size_t)wy * xw + wx];
        Apad[idx] = v;
    }
    __syncthreads();

    const int wave  = tid >> 5;        // 0..7  -> N block (16 columns of out)
    const int lane  = tid & 31;
    const int half  = lane >> 4;       // half-wave select
    const int mn    = lane & 15;       // M index for A-reads, N index for B/C/D
    const int khalf = half * 2;        // A/B VGPR K offset per ISA layout
    const int m0    = half * 8;        // C/D row base per ISA layout

    // ================= Stage A: U_n = Apad(16x16) * Qs[:, 16n:16n+16] ========
    v8f u = {0.f, 0.f, 0.f, 0.f, 0.f, 0.f, 0.f, 0.f};
#pragma unroll
    for (int j = 0; j < 4; ++j) {
        const int k0 = 4 * j + khalf;
        v2f a;  a.x  = Apad[mn * 16 + k0];
                a.y  = Apad[mn * 16 + k0 + 1];
        v2f bb; bb.x = Qs[(k0    ) * 128 + wave * 16 + mn];
                bb.y = Qs[(k0 + 1) * 128 + wave * 16 + mn];
        u = __builtin_amdgcn_wmma_f32_16x16x4_f32(
                false, a, false, bb, (short)0, u, false, false);
    }
    // D-layout -> row-major LDS (wave-private; same-wave DS ordering suffices)
#pragma unroll
    for (int i = 0; i < 8; ++i)
        Ustage[wave][(m0 + i) * 16 + mn] = u[i];

    // ================= Stage B: out[16mt:16mt+16, 16n:16n+16] = P'_mt * U_n ==
    float* outBase = out + (((size_t)b * 4 + si) * (96 * 128)) + wave * 16;
#pragma unroll
    for (int mt = 0; mt < 6; ++mt) {
        v8f d = {0.f, 0.f, 0.f, 0.f, 0.f, 0.f, 0.f, 0.f};
#pragma unroll
        for (int j = 0; j < 4; ++j) {
            const int k0 = 4 * j + khalf;
            v2f a;  a.x  = Ps[(mt * 16 + mn) * 16 + k0];
                    a.y  = Ps[(mt * 16 + mn) * 16 + k0 + 1];
            v2f bb; bb.x = Ustage[wave][(k0    ) * 16 + mn];
                    bb.y = Ustage[wave][(k0 + 1) * 16 + mn];
            d = __builtin_amdgcn_wmma_f32_16x16x4_f32(
                    false, a, false, bb, (short)0, d, false, false);
        }
#pragma unroll
        for (int i = 0; i < 8; ++i)
            outBase[(size_t)(mt * 16 + m0 + i) * 128 + mn] = d[i];
    }
}

extern "C" void kernel_launch(void* const* d_in, const int* in_sizes, int n_in,
                              void* d_out, int out_size, void* d_ws, size_t ws_size,
                              hipStream_t stream) {
    const float* sim0 = (const float*)d_in[0];  // (B,1,11,15)
    const float* sim1 = (const float*)d_in[1];  // (B,1,10,14)
    const float* sim2 = (const float*)d_in[2];  // (B,1, 9,13)
    const float* sim3 = (const float*)d_in[3];  // (B,1, 7,11)
    float* out = (float*)d_out;                 // (B,4,96,128)

    const int B = in_sizes[0] / (11 * 15);

    dim3 grid(B, 4, 1);
    dim3 block(256, 1, 1);
    winscatter_wmma_kernel<<<grid, block, 0, stream>>>(sim0, sim1, sim2, sim3, out, B);
}